// Model_33913061769293
// MI455X (gfx1250) — compile-verified
//
#include <hip/hip_runtime.h>
#include <math.h>

typedef __attribute__((ext_vector_type(2))) float v2f;
typedef __attribute__((ext_vector_type(8))) float v8f;

// ---- workspace layout (float offsets) ----
#define OFF_P0      0        // 16384  mix probs layer0
#define OFF_P1      16384    // 16384
#define OFF_P2      32768    // 16384
#define OFF_P3      49152    // 8192
#define OFF_CF0     57344    // 8192*4 gate coeffs layer0
#define OFF_CF1     90112    // 8192*4
#define OFF_CF2     122880   // 8192*4
#define OFF_CF3     155648   // 4096*4
#define OFF_LOGITS  172032   // 2048*16
// total = 204800 floats = 800 KB

// Gate basis: coefficients of each of the 16 gates on {1, A, B, AB}
__device__ const float GB[16][4] = {
    {0.f, 0.f, 0.f, 0.f}, {0.f, 0.f, 0.f, 1.f}, {0.f, 1.f, 0.f,-1.f}, {0.f, 1.f, 0.f, 0.f},
    {0.f, 0.f, 1.f,-1.f}, {0.f, 0.f, 1.f, 0.f}, {0.f, 1.f, 1.f,-2.f}, {0.f, 1.f, 1.f,-1.f},
    {1.f,-1.f,-1.f, 1.f}, {1.f,-1.f,-1.f, 2.f}, {1.f, 0.f,-1.f, 0.f}, {1.f, 0.f,-1.f, 1.f},
    {1.f,-1.f, 0.f, 0.f}, {1.f,-1.f, 0.f, 1.f}, {1.f, 0.f, 0.f,-1.f}, {1.f, 0.f, 0.f, 0.f}
};

// p[j] = softmax(c[n,:,o])[0] = sigmoid(c[n,0,o] - c[n,1,o]),  n=j/bo, o=j%bo
__global__ __launch_bounds__(256) void mix_prob_kernel(const float* __restrict__ c,
                                                       float* __restrict__ p,
                                                       int n_out, int bo) {
    int j = blockIdx.x * blockDim.x + threadIdx.x;
    if (j >= n_out) return;
    int n = j / bo, o = j - n * bo;
    float a  = c[n * 2 * bo + o];
    float b2 = c[n * 2 * bo + bo + o];
    p[j] = 1.0f / (1.0f + expf(b2 - a));
}

// cf[g] = softmax(w[:,g]) dot GATE_BASIS  -> 4 coefficients per gate
__global__ __launch_bounds__(256) void gate_coeff_kernel(const float* __restrict__ w,
                                                         float4* __restrict__ cf, int G) {
    int g = blockIdx.x * blockDim.x + threadIdx.x;
    if (g >= G) return;
    float wv[16];
    float m = -INFINITY;
#pragma unroll
    for (int k = 0; k < 16; ++k) { wv[k] = w[k * G + g]; m = fmaxf(m, wv[k]); }
    float z = 0.f, a0 = 0.f, a1 = 0.f, a2 = 0.f, a3 = 0.f;
#pragma unroll
    for (int k = 0; k < 16; ++k) {
        float e = expf(wv[k] - m);
        z += e;
        a0 = fmaf(e, GB[k][0], a0);
        a1 = fmaf(e, GB[k][1], a1);
        a2 = fmaf(e, GB[k][2], a2);
        a3 = fmaf(e, GB[k][3], a3);
    }
    float inv = 1.0f / z;
    cf[g] = make_float4(a0 * inv, a1 * inv, a2 * inv, a3 * inv);
}

// Main fused kernel: block = (category cat, batch tile of 16 rows), 256 threads.
// Thread tid owns pair-chain k = cat*256+tid (input features 2k,2k+1), runs all
// 4 layers in registers, deposits out3 into LDS, then the per-category logit
// (sum over 256 features) is computed with chained V_WMMA_F32_16X16X4_F32
// against an all-ones B matrix (f32-exact reduction on the matrix pipe).
#define TILE_STRIDE 258   // padded row stride (words) -> conflict-free A reads

__global__ __launch_bounds__(256) void chain_kernel(const float* __restrict__ X,
                                                    const float* __restrict__ ws,
                                                    float* __restrict__ logits) {
    const int tid = threadIdx.x;
    const int cat = blockIdx.x;   // 0..15
    const int bt  = blockIdx.y;   // 0..127
    const int k   = (cat << 8) | tid;

    // ---- load per-chain weights once (reused across 16 batch rows) ----
    const float4* pb[3]  = { (const float4*)(ws + OFF_P0),
                             (const float4*)(ws + OFF_P1),
                             (const float4*)(ws + OFF_P2) };
    const float4* cb[3]  = { (const float4*)(ws + OFF_CF0),
                             (const float4*)(ws + OFF_CF1),
                             (const float4*)(ws + OFF_CF2) };
    float4 pm[3], cfa[3], cfb[3];
#pragma unroll
    for (int l = 0; l < 3; ++l) {
        pm[l]  = pb[l][k];          // p[4k..4k+3]
        cfa[l] = cb[l][2 * k];      // coeffs of gate 2k
        cfb[l] = cb[l][2 * k + 1];  // coeffs of gate 2k+1
    }
    const float* p3 = ws + OFF_P3;
    float p3a = p3[2 * k], p3b = p3[2 * k + 1];
    float4 cf3 = ((const float4*)(ws + OFF_CF3))[k];

    __shared__ float tile[16 * TILE_STRIDE];

    const size_t colbase = (size_t)cat * 512 + 2 * (size_t)tid;
#pragma unroll 1
    for (int bi = 0; bi < 16; ++bi) {
        size_t row = (size_t)(bt * 16 + bi) * 8192;
        if (bi + 1 < 16)  // warm L2/L0 for the next batch row -> global_prefetch_b8
            __builtin_prefetch(X + row + 8192 + colbase, 0, 0);
        float2 xv = *(const float2*)(X + row + colbase);
        float x0 = xv.x, x1 = xv.y;
#pragma unroll
        for (int l = 0; l < 3; ++l) {
            float d  = x0 - x1;
            float A0 = fmaf(pm[l].x, d, x1);
            float B0 = fmaf(pm[l].y, d, x1);
            float A1 = fmaf(pm[l].z, d, x1);
            float B1 = fmaf(pm[l].w, d, x1);
            // gate: c0 + c1*A + c2*B + c3*A*B == (c3*A + c2)*B + (c1*A + c0)
            float y0 = fmaf(fmaf(cfa[l].w, A0, cfa[l].z), B0, fmaf(cfa[l].y, A0, cfa[l].x));
            float y1 = fmaf(fmaf(cfb[l].w, A1, cfb[l].z), B1, fmaf(cfb[l].y, A1, cfb[l].x));
            x0 = y0; x1 = y1;
        }
        float d  = x0 - x1;
        float A  = fmaf(p3a, d, x1);
        float Bv = fmaf(p3b, d, x1);
        float out = fmaf(fmaf(cf3.w, A, cf3.z), Bv, fmaf(cf3.y, A, cf3.x));
        tile[bi * TILE_STRIDE + tid] = out;
    }
    __syncthreads();

#if __has_builtin(__builtin_amdgcn_wmma_f32_16x16x4_f32)
    // A layout (f32 16x4): lanes 0-15 hold M, VGPR0/1 = K{0,1}; lanes 16-31 = K{2,3}.
    // M = batch row in tile, K = feature chunk. B = all-ones 4x16 -> every D column
    // holds the row sum. All 8 waves execute identically (EXEC all-ones per wave).
    v8f acc = {0.f, 0.f, 0.f, 0.f, 0.f, 0.f, 0.f, 0.f};
    v2f bones = {1.0f, 1.0f};
    const int L  = tid & 15;
    const int hi = (tid >> 4) & 1;
    const float* rowp = tile + L * TILE_STRIDE + 2 * hi;
    for (int c = 0; c < 64; ++c) {
        v2f a = *(const v2f*)(rowp + 4 * c);
        acc = __builtin_amdgcn_wmma_f32_16x16x4_f32(false, a, false, bones,
                                                    (short)0, acc, false, false);
    }
    // D layout: VGPR r -> (M=r, N=lane) lanes0-15 ; (M=8+r, N=lane-16) lanes16-31.
    // Lane 0 (N=0) holds sums for batch rows 0..7, lane 16 for rows 8..15.
    if (tid == 0 || tid == 16) {
        int base = (tid >> 4) * 8;
#pragma unroll
        for (int r = 0; r < 8; ++r)
            logits[(size_t)(bt * 16 + base + r) * 16 + cat] = acc[r];
    }
#else
    if (tid < 16) {
        float s = 0.f;
        for (int g = 0; g < 256; ++g) s += tile[tid * TILE_STRIDE + g];
        logits[(size_t)(bt * 16 + tid) * 16 + cat] = s;
    }
#endif
}

__global__ __launch_bounds__(256) void softmax16_kernel(const float* __restrict__ logits,
                                                        float* __restrict__ out) {
    int b = blockIdx.x * blockDim.x + threadIdx.x;
    if (b >= 2048) return;
    float v[16];
    float m = -INFINITY;
#pragma unroll
    for (int j = 0; j < 16; ++j) { v[j] = logits[b * 16 + j]; m = fmaxf(m, v[j]); }
    float z = 0.f;
#pragma unroll
    for (int j = 0; j < 16; ++j) { v[j] = expf(v[j] - m); z += v[j]; }
    float inv = 1.0f / z;
#pragma unroll
    for (int j = 0; j < 16; ++j) out[b * 16 + j] = v[j] * inv;
}

extern "C" void kernel_launch(void* const* d_in, const int* in_sizes, int n_in,
                              void* d_out, int out_size, void* d_ws, size_t ws_size,
                              hipStream_t stream) {
    const float* X  = (const float*)d_in[0];
    const float* c0 = (const float*)d_in[1];
    const float* w0 = (const float*)d_in[2];
    const float* c1 = (const float*)d_in[3];
    const float* w1 = (const float*)d_in[4];
    const float* c2 = (const float*)d_in[5];
    const float* w2 = (const float*)d_in[6];
    const float* c3 = (const float*)d_in[7];
    const float* w3 = (const float*)d_in[8];
    float* ws  = (float*)d_ws;
    float* out = (float*)d_out;

    // weight preprocessing (tiny)
    mix_prob_kernel<<<64, 256, 0, stream>>>(c0, ws + OFF_P0, 16384, 4);
    mix_prob_kernel<<<64, 256, 0, stream>>>(c1, ws + OFF_P1, 16384, 4);
    mix_prob_kernel<<<64, 256, 0, stream>>>(c2, ws + OFF_P2, 16384, 4);
    mix_prob_kernel<<<32, 256, 0, stream>>>(c3, ws + OFF_P3, 8192, 2);
    gate_coeff_kernel<<<32, 256, 0, stream>>>(w0, (float4*)(ws + OFF_CF0), 8192);
    gate_coeff_kernel<<<32, 256, 0, stream>>>(w1, (float4*)(ws + OFF_CF1), 8192);
    gate_coeff_kernel<<<32, 256, 0, stream>>>(w2, (float4*)(ws + OFF_CF2), 8192);
    gate_coeff_kernel<<<16, 256, 0, stream>>>(w3, (float4*)(ws + OFF_CF3), 4096);

    // fused 4-layer chain + WMMA logit reduction (reads X exactly once: 64 MB)
    dim3 grid(16, 128);
    chain_kernel<<<grid, 256, 0, stream>>>(X, ws, ws + OFF_LOGITS);

    // final 16-way softmax per batch row
    softmax16_kernel<<<8, 256, 0, stream>>>(ws + OFF_LOGITS, out);
}